// write_head_5454608466465
// MI455X (gfx1250) — compile-verified
//
#include <hip/hip_runtime.h>
#include <math.h>

typedef __attribute__((ext_vector_type(2))) float v2f;
typedef __attribute__((ext_vector_type(8))) float v8f;

#define NN    16384
#define MD    512
#define CTRLD 1024
#define OUTF  1542           // 3*MD + 6
#define EPSF  1e-16f

// ---- workspace layout (float indices) ----
#define WS_SUMU   0
#define WS_SUMWP  1
#define WS_BETA   8
#define WS_G      9
#define WS_S0     10          // s0,s1,s2 at 10,11,12
#define WS_GAMMA  13
#define WS_KNORM  14
#define WS_O      64          // o[1542]
#define WS_U      2048        // u[N]   = exp(beta*sim)
#define WS_WG     (2048 + NN) // wg[N]
#define WS_WP     (2048 + 2*NN) // wp[N]

__device__ __forceinline__ float softplus_f(float x) {
  return (x > 20.f) ? x : log1pf(__expf(x));
}

// K1: o[row] = dot(W[row,:], emb) + b[row]   (1542 blocks x 256 thr)
__global__ void ntm_o_kernel(const float* __restrict__ emb,
                             const float* __restrict__ W,
                             const float* __restrict__ b,
                             float* __restrict__ ws) {
  __shared__ float red[256];
  const int row = blockIdx.x;
  const float* wr = W + (size_t)row * CTRLD;
  float s = 0.f;
  for (int t = threadIdx.x; t < CTRLD; t += 256)
    s = fmaf(wr[t], emb[t], s);
  red[threadIdx.x] = s;
  __syncthreads();
  for (int off = 128; off > 0; off >>= 1) {
    if (threadIdx.x < off) red[threadIdx.x] += red[threadIdx.x + off];
    __syncthreads();
  }
  if (threadIdx.x == 0) ws[WS_O + row] = red[0] + b[row];
}

// K2: scalar gates + ||k|| ; zero the global accumulators (1 block x 256)
__global__ void ntm_scalars_kernel(float* __restrict__ ws) {
  __shared__ float red[256];
  const float* o = ws + WS_O;
  const float x0 = o[threadIdx.x];
  const float x1 = o[threadIdx.x + 256];
  red[threadIdx.x] = x0 * x0 + x1 * x1;   // k = o[0:512]
  __syncthreads();
  for (int off = 128; off > 0; off >>= 1) {
    if (threadIdx.x < off) red[threadIdx.x] += red[threadIdx.x + off];
    __syncthreads();
  }
  if (threadIdx.x == 0) {
    ws[WS_KNORM] = sqrtf(red[0]);
    ws[WS_BETA]  = softplus_f(o[MD]);
    ws[WS_G]     = 1.f / (1.f + __expf(-o[MD + 1]));
    const float a0 = o[MD + 2], a1 = o[MD + 3], a2 = o[MD + 4];
    const float mx = fmaxf(a0, fmaxf(a1, a2));
    const float e0 = __expf(a0 - mx), e1 = __expf(a1 - mx), e2 = __expf(a2 - mx);
    const float inv = 1.f / (e0 + e1 + e2);
    ws[WS_S0 + 0] = e0 * inv;
    ws[WS_S0 + 1] = e1 * inv;
    ws[WS_S0 + 2] = e2 * inv;
    ws[WS_GAMMA]  = 1.f + softplus_f(o[MD + 5]);
    ws[WS_SUMU]   = 0.f;
    ws[WS_SUMWP]  = 0.f;
  }
}

// K3: cosine similarity via V_WMMA_F32_16X16X4_F32.
// One wave per 16-row tile of memory; A = 16x4 fp32 tile, B = key slice
// broadcast over all 16 columns => every column of D holds the 16 dots.
// Row norms accumulated in the same pass (same loads).  128 blocks x 256.
__global__ void ntm_sim_kernel(const float* __restrict__ memory,
                               float* __restrict__ ws) {
  __shared__ float wsum[8];
  const int tid  = threadIdx.x;
  const int wave = tid >> 5;
  const int lane = tid & 31;
  const int m    = lane & 15;       // row within tile (both halves)
  const int half = lane >> 4;       // K-phase: 0 -> K%4 in {0,1}, 1 -> {2,3}
  const int koff = half * 2;
  const int tileRow = (blockIdx.x * 8 + wave) * 16;

  const float* rowp = memory + (size_t)(tileRow + m) * MD;
  const float* kvec = ws + WS_O;    // k = o[0:512]

  v8f acc = {};
  float nsq = 0.f;
  for (int k0 = 0; k0 < MD; k0 += 4) {
    const float2 av = *(const float2*)(rowp + k0 + koff);
    const float2 bv = *(const float2*)(kvec + k0 + koff);
    v2f a;  a[0] = av.x;  a[1] = av.y;
    v2f bb; bb[0] = bv.x; bb[1] = bv.y;
    // D = A(16x4) x B(4x16) + C ; 8 args: neg_a,A,neg_b,B,c_mod,C,reuse_a,reuse_b
    acc = __builtin_amdgcn_wmma_f32_16x16x4_f32(
        false, a, false, bb, (short)0, acc, false, false);
    nsq = fmaf(av.x, av.x, nsq);
    nsq = fmaf(av.y, av.y, nsq);
  }
  // combine K-phase partial norms between lane l and l^16 (same row m)
  nsq += __shfl_xor(nsq, 16, 32);

  // Extract dot for row m from C layout: VGPR r holds M = half*8 + r.
  const int idx = lane & 7;
  float e_ = acc[0];
  e_ = (idx == 1) ? acc[1] : e_;
  e_ = (idx == 2) ? acc[2] : e_;
  e_ = (idx == 3) ? acc[3] : e_;
  e_ = (idx == 4) ? acc[4] : e_;
  e_ = (idx == 5) ? acc[5] : e_;
  e_ = (idx == 6) ? acc[6] : e_;
  e_ = (idx == 7) ? acc[7] : e_;
  const float sw = __shfl_xor(e_, 16, 32);
  const float dot = (((lane >> 3) & 1) == half) ? e_ : sw;

  const float beta  = ws[WS_BETA];
  const float knorm = ws[WS_KNORM];
  const float sim = dot / (knorm * sqrtf(nsq) + EPSF);
  const float u = __expf(beta * sim);   // softmax numerator (beta*sim bounded)
  if (lane < 16) ws[WS_U + tileRow + m] = u;

  // block-sum of u (count each row once: lanes 0..15 only)
  float c = (lane < 16) ? u : 0.f;
  for (int off = 16; off > 0; off >>= 1) c += __shfl_down(c, off, 32);
  if (lane == 0) wsum[wave] = c;
  __syncthreads();
  if (tid == 0) {
    float t = 0.f;
    for (int i = 0; i < 8; ++i) t += wsum[i];
    atomicAdd(ws + WS_SUMU, t);
  }
}

// K4: wg = g*wc + (1-g)*w_prev   (N/256 blocks x 256)
__global__ void ntm_wg_kernel(const float* __restrict__ w_prev,
                              float* __restrict__ ws) {
  const int i = blockIdx.x * 256 + threadIdx.x;
  const float g    = ws[WS_G];
  const float invS = 1.f / ws[WS_SUMU];
  const float wc   = ws[WS_U + i] * invS;
  ws[WS_WG + i] = fmaf(g, wc, (1.f - g) * w_prev[i]);
}

// K5: circular shift + sharpen, accumulate sum(wp)   (N/256 blocks x 256)
__global__ void ntm_shift_kernel(float* __restrict__ ws) {
  __shared__ float red[256];
  const int i = blockIdx.x * 256 + threadIdx.x;
  const float s0 = ws[WS_S0 + 0];
  const float s1 = ws[WS_S0 + 1];
  const float s2 = ws[WS_S0 + 2];
  const float gamma = ws[WS_GAMMA];
  const int im = (i == 0) ? (NN - 1) : (i - 1);
  const int ip = (i == NN - 1) ? 0 : (i + 1);
  // roll(wg,+1)[i] = wg[i-1] ; roll(wg,-1)[i] = wg[i+1]
  const float wsv = s0 * ws[WS_WG + im] + s1 * ws[WS_WG + i] + s2 * ws[WS_WG + ip];
  const float wp = powf(wsv + EPSF, gamma);
  ws[WS_WP + i] = wp;
  red[threadIdx.x] = wp;
  __syncthreads();
  for (int off = 128; off > 0; off >>= 1) {
    if (threadIdx.x < off) red[threadIdx.x] += red[threadIdx.x + off];
    __syncthreads();
  }
  if (threadIdx.x == 0) atomicAdd(ws + WS_SUMWP, red[0]);
}

// K6: w out + rank-1 memory update, float4 streaming. (N blocks x 128)
__global__ void ntm_update_kernel(const float* __restrict__ memory,
                                  const float* __restrict__ ws,
                                  float* __restrict__ out) {
  const int row = blockIdx.x;
  const float w = ws[WS_WP + row] / ws[WS_SUMWP];
  if (threadIdx.x == 0) out[row] = w;

  const float* o = ws + WS_O;
  const int j = threadIdx.x * 4;
  const float4 mv = *(const float4*)(memory + (size_t)row * MD + j);
  const float e0 = o[MD + 6 + j],       e1 = o[MD + 7 + j];
  const float e2 = o[MD + 8 + j],       e3 = o[MD + 9 + j];
  const float a0 = o[2 * MD + 6 + j],   a1 = o[2 * MD + 7 + j];
  const float a2 = o[2 * MD + 8 + j],   a3 = o[2 * MD + 9 + j];
  float4 r;
  r.x = fmaf(mv.x, 1.f - w * e0, w * a0);
  r.y = fmaf(mv.y, 1.f - w * e1, w * a1);
  r.z = fmaf(mv.z, 1.f - w * e2, w * a2);
  r.w = fmaf(mv.w, 1.f - w * e3, w * a3);
  *(float4*)(out + NN + (size_t)row * MD + j) = r;
}

extern "C" void kernel_launch(void* const* d_in, const int* in_sizes, int n_in,
                              void* d_out, int out_size, void* d_ws, size_t ws_size,
                              hipStream_t stream) {
  const float* emb    = (const float*)d_in[0]; // (1,1024)
  const float* w_prev = (const float*)d_in[1]; // (1,16384)
  const float* memory = (const float*)d_in[2]; // (16384,512)
  const float* W      = (const float*)d_in[3]; // (1542,1024)
  const float* b      = (const float*)d_in[4]; // (1542,)
  float* out = (float*)d_out;                  // [w (16384) | new_memory (16384*512)]
  float* ws  = (float*)d_ws;

  ntm_o_kernel      <<<OUTF,      256, 0, stream>>>(emb, W, b, ws);
  ntm_scalars_kernel<<<1,         256, 0, stream>>>(ws);
  ntm_sim_kernel    <<<NN / 128,  256, 0, stream>>>(memory, ws);
  ntm_wg_kernel     <<<NN / 256,  256, 0, stream>>>(w_prev, ws);
  ntm_shift_kernel  <<<NN / 256,  256, 0, stream>>>(ws);
  ntm_update_kernel <<<NN,        128, 0, stream>>>(memory, ws, out);
}